// EctLayer_54528904790172
// MI455X (gfx1250) — compile-verified
//
#include <hip/hip_runtime.h>

typedef float v2f __attribute__((ext_vector_type(2)));
typedef float v8f __attribute__((ext_vector_type(8)));

#define NODES   200000
#define FDIM    3
#define TDIM    32
#define SDIM    32
#define NSEG    128
#define TILE    16
#define NTILES  (NODES / TILE)          // 12500, exact
#define BLOCKS  256
#define THREADS 256
#define NWAVES  ((BLOCKS * THREADS) / 32)               // 2048 waves
#define TPW     ((NTILES + NWAVES - 1) / NWAVES)        // 7 tiles per wave

__global__ __launch_bounds__(256) void ect_zero(float* __restrict__ out, int n) {
    int i = blockIdx.x * blockDim.x + threadIdx.x;
    if (i < n) out[i] = 0.0f;
}

__global__ __launch_bounds__(THREADS) void ect_fused(
    const float* __restrict__ x,      // [N, 3]
    const int*   __restrict__ idx,    // [N], sorted
    const float* __restrict__ v,      // [3, 32]
    const float* __restrict__ lin,    // [32]
    const int*   __restrict__ scale_p,// [1]
    float*       __restrict__ out)    // [128, 32, 32]
{
    const int lane = threadIdx.x & 31;
    const int gw   = (blockIdx.x * blockDim.x + threadIdx.x) >> 5;
    const int tile0 = gw * TPW;
    const int tile1 = (tile0 + TPW < NTILES) ? (tile0 + TPW) : NTILES;
    if (tile0 >= tile1) return;

    const float scale = (float)scale_p[0];

    // Uniform thresholds: scale * lin[s]  (scalarized loads -> SGPRs; the
    // compiler folds these into v_fma_f32 (c*scale - slin) directly).
    float slin[SDIM];
#pragma unroll
    for (int s = 0; s < SDIM; ++s) slin[s] = scale * lin[s];

    const bool hi   = (lane >= 16);
    const int  nsub = lane & 15;

    // B operand (4x16 f32) for each 16-wide half of v.
    // Layout: VGPR0 = row K=0 (lanes 0-15) / K=2 (lanes 16-31); VGPR1 = K=1 / K=3(=0).
    v2f bA, bB;
    {
        const int tA = nsub, tB = nsub + 16;
        bA.x = hi ? v[2 * TDIM + tA] : v[0 * TDIM + tA];
        bA.y = hi ? 0.0f             : v[1 * TDIM + tA];
        bB.x = hi ? v[2 * TDIM + tB] : v[0 * TDIM + tB];
        bB.y = hi ? 0.0f             : v[1 * TDIM + tB];
    }

    // Per-lane segment accumulators: this lane owns t = nsub and t = nsub+16.
    float acc0[SDIM], acc1[SDIM];
#pragma unroll
    for (int s = 0; s < SDIM; ++s) { acc0[s] = 0.0f; acc1[s] = 0.0f; }
    int  cur_b = idx[tile0 * TILE];
    bool dirty = false;

    // Per-lane offset into a row of x for the A.x element: 0 (lo lanes) or 2 (hi lanes).
    const int kxoff = hi ? 2 : 0;

    for (int tl = tile0; tl < tile1; ++tl) {
        const int base = tl * TILE;
        const int b0   = idx[base];
        const int b15  = idx[base + 15];

        if (b0 != cur_b) {
            if (dirty) {
#pragma unroll
                for (int s = 0; s < SDIM; ++s) {
                    unsafeAtomicAdd(out + (cur_b * SDIM + s) * TDIM + nsub,      acc0[s]);
                    unsafeAtomicAdd(out + (cur_b * SDIM + s) * TDIM + nsub + 16, acc1[s]);
                    acc0[s] = 0.0f; acc1[s] = 0.0f;
                }
                dirty = false;
            }
            cur_b = b0;
        }

        // Prefetch next tile's x rows into the near (WGP) cache level.
        __builtin_prefetch(x + (size_t)(base + TILE) * FDIM, 0, 3);

        // A operand (16x4 f32): lane L -> M = L&15; K = {0,1} (lo lanes) / {2,3} (hi lanes).
        // Both loads are unconditional (always in-bounds) -> no EXEC-mask branch;
        // the K=3 zero pad is a post-load v_cndmask.
        const float* xr = x + (size_t)(base + nsub) * FDIM;
        const float ax = xr[kxoff];   // K=0 (lo) / K=2 (hi)
        const float ay = xr[1];       // K=1 (lo) / discarded (hi)
        v2f a;
        a.x = ax;
        a.y = hi ? 0.0f : ay;

        // nh tile: 16 nodes x 32 t via two 16x16x4 f32 WMMAs.
        v8f c0 = {}, c1 = {};
        c0 = __builtin_amdgcn_wmma_f32_16x16x4_f32(false, a, false, bA, (short)0, c0, false, false);
        c1 = __builtin_amdgcn_wmma_f32_16x16x4_f32(false, a, false, bB, (short)0, c1, false, false);

        if (b0 == b15) {
            // Uniform tile: all 16 nodes in segment cur_b; accumulate in registers.
#pragma unroll
            for (int r = 0; r < 8; ++r) {
                const float snh0 = scale * c0[r];   // row M = r (lo lanes) / r+8 (hi lanes)
                const float snh1 = scale * c1[r];
#pragma unroll
                for (int s = 0; s < SDIM; ++s) {
                    const float e0 = __expf(snh0 - slin[s]);  // exp(-z), z = slin - snh
                    const float e1 = __expf(snh1 - slin[s]);
                    acc0[s] += __builtin_amdgcn_rcpf(1.0f + e0);
                    acc1[s] += __builtin_amdgcn_rcpf(1.0f + e1);
                }
            }
            dirty = true;
        } else {
            // Boundary tile (rare, ~127 total): per-row direct atomics.
#pragma unroll
            for (int r = 0; r < 8; ++r) {
                const int m    = hi ? (r + 8) : r;
                const int bseg = idx[base + m];
                const float snh0 = scale * c0[r];
                const float snh1 = scale * c1[r];
#pragma unroll
                for (int s = 0; s < SDIM; ++s) {
                    const float e0 = __expf(snh0 - slin[s]);
                    const float e1 = __expf(snh1 - slin[s]);
                    unsafeAtomicAdd(out + (bseg * SDIM + s) * TDIM + nsub,
                                    __builtin_amdgcn_rcpf(1.0f + e0));
                    unsafeAtomicAdd(out + (bseg * SDIM + s) * TDIM + nsub + 16,
                                    __builtin_amdgcn_rcpf(1.0f + e1));
                }
            }
        }
    }

    if (dirty) {
#pragma unroll
        for (int s = 0; s < SDIM; ++s) {
            unsafeAtomicAdd(out + (cur_b * SDIM + s) * TDIM + nsub,      acc0[s]);
            unsafeAtomicAdd(out + (cur_b * SDIM + s) * TDIM + nsub + 16, acc1[s]);
        }
    }
}

extern "C" void kernel_launch(void* const* d_in, const int* in_sizes, int n_in,
                              void* d_out, int out_size, void* d_ws, size_t ws_size,
                              hipStream_t stream) {
    const float* x     = (const float*)d_in[0];   // [200000, 3]
    const int*   idx   = (const int*)d_in[1];     // [200000]
    const float* v     = (const float*)d_in[2];   // [3, 32]
    const float* lin   = (const float*)d_in[3];   // [32]
    const int*   scale = (const int*)d_in[4];     // [1] == 500
    float*       out   = (float*)d_out;           // [128, 32, 32]

    // Output buffer is poisoned by the harness; zero it first (same stream).
    const int zthreads = 256;
    const int zblocks  = (out_size + zthreads - 1) / zthreads;
    ect_zero<<<zblocks, zthreads, 0, stream>>>(out, out_size);

    ect_fused<<<BLOCKS, THREADS, 0, stream>>>(x, idx, v, lin, scale, out);
}